// vssgp_2413771620431
// MI455X (gfx1250) — compile-verified
//
#include <hip/hip_runtime.h>
#include <math.h>

// Problem constants (from reference: N=8192, D=8, K=64, Q=16)
static constexpr int Nn = 8192;
static constexpr int Dd = 8;
static constexpr int Kk = 64;
static constexpr int Qq = 16;
static constexpr int KQ = Kk * Qq;          // 1024
static constexpr int SPLITK = 4;            // deterministic split-K factor
static constexpr float PI_F = 3.14159265358979323846f;

typedef float v2f __attribute__((ext_vector_type(2)));
typedef float v4f __attribute__((ext_vector_type(4)));
typedef float v8f __attribute__((ext_vector_type(8)));

// ---------------------------------------------------------------------------
// Stage 1: elementwise E_phi / E_cos_sq
// One thread per (n, kq); kq fastest so z/var/alpha loads are coalesced and
// X[n][d] is block-uniform (scalarized by the compiler).
// ---------------------------------------------------------------------------
__global__ __launch_bounds__(256) void ephi_kernel(
    const float* __restrict__ X, const float* __restrict__ z,
    const float* __restrict__ weight, const float* __restrict__ mu,
    const float* __restrict__ stdv, const float* __restrict__ alpha,
    const float* __restrict__ var_mu_w, const float* __restrict__ var_sigma_w,
    float* __restrict__ Ephi, float* __restrict__ Ecos) {
  const int idx = blockIdx.x * 256 + threadIdx.x;   // n*KQ + kq
  const int n  = idx >> 10;        // /KQ
  const int kq = idx & (KQ - 1);
  const int q  = kq & (Qq - 1);

  float phase = 0.0f;
  float svar  = 0.0f;
#pragma unroll
  for (int d = 0; d < Dd; ++d) {
    const float x    = X[n * Dd + d];
    const float zz   = z[d * KQ + kq];
    const float mp   = 1.0f / (mu[d * Qq + q] + 1e-8f);
    const float sp   = 1.0f / (2.0f * PI_F * stdv[d * Qq + q] + 1e-8f);
    const float Ew   = fmaf(sp, var_mu_w[d * KQ + kq], mp);
    const float xb   = 2.0f * PI_F * (x - zz);
    phase = fmaf(xb, Ew, phase);
    const float t = sp * xb;
    svar = fmaf(t * t, var_sigma_w[d * KQ + kq], svar);
  }
  const float decay = expf(-0.5f * svar);
  const float a     = alpha[kq];
  const float tok   = 2.0f * weight[q] * (1.0f / (float)Kk);
  const float ephi  = sqrtf(tok) * decay * cosf(a + phase);
  const float d2 = decay * decay;
  const float d4 = d2 * d2;
  const float ecos = tok * fmaf(0.5f * d4, cosf(2.0f * a + 2.0f * phase), 0.5f);

  Ephi[idx] = ephi;
  Ecos[idx] = ecos;
}

// ---------------------------------------------------------------------------
// Stage 2: G = E_phi^T * E_phi  via V_WMMA_F32_16X16X4_F32.
// Block = 128 threads (4 wave32 waves) computing a 64(row) x 128(col) tile.
// Wave (wy,wx in 2x2) owns a 32x64 sub-tile = 2x4 WMMA accumulators.
// gridDim.z = split-K segments (fixed n-ranges -> deterministic); each
// segment writes to out + z*splitStride (partial G in workspace, or G itself
// when splitStride==0 and gridDim.z==1).
// ---------------------------------------------------------------------------
__global__ __launch_bounds__(128) void gram_wmma_kernel(
    const float* __restrict__ E, float* __restrict__ out,
    int nPerSplit, size_t splitStride) {
  __shared__ float As[16][64];    // [n within chunk][i within 64-row tile]
  __shared__ float Bs[16][128];   // [n within chunk][j within 128-col tile]

  const int iBase = blockIdx.y * 64;
  const int jBase = blockIdx.x * 128;
  float* G = out + (size_t)blockIdx.z * splitStride;
  const int nBegin = blockIdx.z * nPerSplit;
  const int nEnd   = nBegin + nPerSplit;

  const int t     = threadIdx.x;
  const int wave  = t >> 5;
  const int lane  = t & 31;
  const int half  = lane >> 4;    // 0: lanes 0-15, 1: lanes 16-31
  const int lr    = lane & 15;
  const int wy    = wave & 1;     // 2 row groups of 32
  const int wx    = wave >> 1;    // 2 col groups of 64

  v8f acc[2][4];
#pragma unroll
  for (int m = 0; m < 2; ++m)
#pragma unroll
    for (int nn = 0; nn < 4; ++nn) acc[m][nn] = (v8f)0.0f;

  // Global-load mapping (128 threads):
  //   A tile: 16 rows x 64  floats ->  8 floats/thread (2 x v4f)
  //   B tile: 16 rows x 128 floats -> 16 floats/thread (4 x v4f)
  const int lrow  = t >> 3;           // 0..15
  const int lcolA = (t & 7) * 8;      // 0..56  step 8
  const int lcolB = (t & 7) * 16;     // 0..112 step 16

  for (int n0 = nBegin; n0 < nEnd; n0 += 16) {
    const float* src = E + (size_t)(n0 + lrow) * KQ;
    const v4f a0 = *(const v4f*)(src + iBase + lcolA);
    const v4f a1 = *(const v4f*)(src + iBase + lcolA + 4);
    const v4f b0 = *(const v4f*)(src + jBase + lcolB);
    const v4f b1 = *(const v4f*)(src + jBase + lcolB + 4);
    const v4f b2 = *(const v4f*)(src + jBase + lcolB + 8);
    const v4f b3 = *(const v4f*)(src + jBase + lcolB + 12);
    if (n0 + 16 < nEnd) {
      __builtin_prefetch(src + 16 * KQ + iBase + lcolA, 0, 1);
      __builtin_prefetch(src + 16 * KQ + jBase + lcolB, 0, 1);
    }
    __syncthreads();                       // previous chunk's compute done
    *(v4f*)&As[lrow][lcolA]      = a0;
    *(v4f*)&As[lrow][lcolA + 4]  = a1;
    *(v4f*)&Bs[lrow][lcolB]      = b0;
    *(v4f*)&Bs[lrow][lcolB + 4]  = b1;
    *(v4f*)&Bs[lrow][lcolB + 8]  = b2;
    *(v4f*)&Bs[lrow][lcolB + 12] = b3;
    __syncthreads();                       // LDS tile visible

#pragma unroll
    for (int kk = 0; kk < 16; kk += 4) {
      // A fragment (16x4, M=i, K=n): lane<16 holds K=kk,kk+1; lane>=16 K=kk+2,kk+3
      v2f af[2];
#pragma unroll
      for (int m = 0; m < 2; ++m) {
        const int i = wy * 32 + m * 16 + lr;
        af[m].x = As[kk + half * 2 + 0][i];
        af[m].y = As[kk + half * 2 + 1][i];
      }
      // B fragment (4x16, K=n, N=j): same half-split over K
      v2f bf[4];
#pragma unroll
      for (int nn = 0; nn < 4; ++nn) {
        const int j = wx * 64 + nn * 16 + lr;
        bf[nn].x = Bs[kk + half * 2 + 0][j];
        bf[nn].y = Bs[kk + half * 2 + 1][j];
      }
#pragma unroll
      for (int m = 0; m < 2; ++m)
#pragma unroll
        for (int nn = 0; nn < 4; ++nn)
          acc[m][nn] = __builtin_amdgcn_wmma_f32_16x16x4_f32(
              /*neg_a=*/false, af[m], /*neg_b=*/false, bf[nn],
              /*c_mod=*/(short)0, acc[m][nn],
              /*reuse_a=*/false, /*reuse_b=*/false);
    }
  }

  // Store: C/D 16x16 f32 layout — VGPR v: row v (lanes 0-15) / row v+8 (16-31)
#pragma unroll
  for (int m = 0; m < 2; ++m) {
#pragma unroll
    for (int nn = 0; nn < 4; ++nn) {
      const int row0 = iBase + wy * 32 + m * 16 + half * 8;
      const int col  = jBase + wx * 64 + nn * 16 + lr;
#pragma unroll
      for (int v = 0; v < 8; ++v)
        G[(size_t)(row0 + v) * KQ + col] = acc[m][nn][v];
    }
  }
}

// Sum the SPLITK partial Gram matrices (fixed order -> deterministic).
__global__ __launch_bounds__(256) void reduce_split_kernel(
    const float* __restrict__ Gp, float* __restrict__ G) {
  const size_t i = ((size_t)blockIdx.x * 256 + threadIdx.x) * 4;
  v4f s = *(const v4f*)(Gp + i);
#pragma unroll
  for (int zz = 1; zz < SPLITK; ++zz) {
    const v4f p = *(const v4f*)(Gp + (size_t)zz * KQ * KQ + i);
    s.x += p.x; s.y += p.y; s.z += p.z; s.w += p.w;
  }
  *(v4f*)(G + i) = s;
}

// ---------------------------------------------------------------------------
// Stage 3: diag(G) <- sum_n E_cos_sq[n, :]   (deterministic two-stage)
// ---------------------------------------------------------------------------
__global__ __launch_bounds__(256) void colsum_partial_kernel(
    const float* __restrict__ Ecos, float* __restrict__ partial) {
  const int col = blockIdx.x * 256 + threadIdx.x;   // gridDim.x = KQ/256
  const int seg = blockIdx.y;                       // 16 segments of 512 rows
  float s = 0.0f;
  const int n0 = seg * (Nn / 16);
  for (int n = n0; n < n0 + (Nn / 16); ++n) s += Ecos[(size_t)n * KQ + col];
  partial[seg * KQ + col] = s;
}

__global__ __launch_bounds__(256) void diag_fix_kernel(
    const float* __restrict__ partial, float* __restrict__ G) {
  const int col = blockIdx.x * 256 + threadIdx.x;
  float s = 0.0f;
#pragma unroll
  for (int i = 0; i < 16; ++i) s += partial[i * KQ + col];
  G[(size_t)col * KQ + col] = s;
}

__global__ __launch_bounds__(256) void diag_direct_kernel(
    const float* __restrict__ Ecos, float* __restrict__ G) {
  const int col = blockIdx.x * 256 + threadIdx.x;
  float s = 0.0f;
  for (int n = 0; n < Nn; ++n) s += Ecos[(size_t)n * KQ + col];
  G[(size_t)col * KQ + col] = s;
}

// ---------------------------------------------------------------------------
extern "C" void kernel_launch(void* const* d_in, const int* in_sizes, int n_in,
                              void* d_out, int out_size, void* d_ws, size_t ws_size,
                              hipStream_t stream) {
  const float* X     = (const float*)d_in[0];
  const float* z     = (const float*)d_in[1];
  const float* wgt   = (const float*)d_in[2];
  const float* mu    = (const float*)d_in[3];
  const float* stdv  = (const float*)d_in[4];
  const float* alpha = (const float*)d_in[5];
  const float* vmw   = (const float*)d_in[6];
  const float* vsw   = (const float*)d_in[7];

  float* out   = (float*)d_out;
  float* Ephi  = out;                                     // [N, KQ]
  float* G     = out + (size_t)Nn * KQ;                   // [KQ, KQ]
  float* Ecos  = out + (size_t)Nn * KQ + (size_t)KQ * KQ; // [N, KQ]

  // Stage 1: elementwise
  ephi_kernel<<<(Nn * KQ) / 256, 256, 0, stream>>>(
      X, z, wgt, mu, stdv, alpha, vmw, vsw, Ephi, Ecos);

  // Stage 2: FP32 WMMA Gram matrix (split-K if workspace allows)
  const size_t splitBytes = (size_t)SPLITK * KQ * KQ * sizeof(float);
  const size_t diagBytes  = (size_t)16 * KQ * sizeof(float);
  const bool useSplit = ws_size >= splitBytes + diagBytes;

  dim3 ggrid(KQ / 128, KQ / 64, useSplit ? SPLITK : 1);
  if (useSplit) {
    float* Gpart = (float*)d_ws;
    gram_wmma_kernel<<<ggrid, 128, 0, stream>>>(
        Ephi, Gpart, Nn / SPLITK, (size_t)KQ * KQ);
    reduce_split_kernel<<<(KQ * KQ) / (256 * 4), 256, 0, stream>>>(Gpart, G);
  } else {
    gram_wmma_kernel<<<ggrid, 128, 0, stream>>>(Ephi, G, Nn, 0);
  }

  // Stage 3: diagonal replacement
  if (ws_size >= (useSplit ? splitBytes + diagBytes : diagBytes)) {
    float* partial = (float*)d_ws + (useSplit ? (size_t)SPLITK * KQ * KQ : 0);
    colsum_partial_kernel<<<dim3(KQ / 256, 16), 256, 0, stream>>>(Ecos, partial);
    diag_fix_kernel<<<KQ / 256, 256, 0, stream>>>(partial, G);
  } else {
    diag_direct_kernel<<<KQ / 256, 256, 0, stream>>>(Ecos, G);
  }
}